// EvoformerBlock_4337916970069
// MI455X (gfx1250) — compile-verified
//
#include <hip/hip_runtime.h>
#include <math.h>

// ---------------- CDNA5 WMMA f32 16x16x4 ----------------
typedef __attribute__((ext_vector_type(2))) float v2f;
typedef __attribute__((ext_vector_type(8))) float v8f;

#define LSEQ   384
#define DMODEL 256
#define DPAIR  128
#define DHID   32
#define DFF    1024

__device__ __forceinline__ v8f wmma_f32(v2f a, v2f b, v8f c) {
  // D = A(16x4,f32) * B(4x16,f32) + C(16x16,f32)
  return __builtin_amdgcn_wmma_f32_16x16x4_f32(false, a, false, b, (short)0, c,
                                               false, false);
}

__device__ __forceinline__ float gelu_exact(float x) {
  return 0.5f * x * (1.0f + erff(x * 0.7071067811865475f));
}

// Async global->LDS copy of 16 bytes per lane (GLOBAL_LOAD_ASYNC_TO_LDS_B128,
// ASYNCcnt-tracked). Inline asm: portable across ROCm7.2 / amdgpu-toolchain.
__device__ __forceinline__ void async_copy_b128(void* lds_dst, const void* gsrc) {
  unsigned lds_off = (unsigned)(size_t)lds_dst;  // LDS aperture: addr[31:0] = LDS offset
  asm volatile("global_load_async_to_lds_b128 %0, %1, off"
               :: "v"(lds_off), "v"(gsrc) : "memory");
}
__device__ __forceinline__ void async_wait0() {
  asm volatile("s_wait_asynccnt 0x0" ::: "memory");
}

// ---------------- LayerNorm: one block (256 threads) per row ----------------
__global__ void ln_kernel(const float* __restrict__ x, const float* __restrict__ g,
                          const float* __restrict__ b, float* __restrict__ out) {
  int row = blockIdx.x, tid = threadIdx.x;
  __shared__ float red[DMODEL];
  float v = x[row * DMODEL + tid];
  red[tid] = v;
  __syncthreads();
  for (int s = DMODEL / 2; s > 0; s >>= 1) {
    if (tid < s) red[tid] += red[tid + s];
    __syncthreads();
  }
  float mu = red[0] * (1.0f / DMODEL);
  __syncthreads();
  float d = v - mu;
  red[tid] = d * d;
  __syncthreads();
  for (int s = DMODEL / 2; s > 0; s >>= 1) {
    if (tid < s) red[tid] += red[tid + s];
    __syncthreads();
  }
  float var = red[0] * (1.0f / DMODEL);
  out[row * DMODEL + tid] = d * rsqrtf(var + 1e-5f) * g[tid] + b[tid];
}

// ---------------- Generic WMMA GEMM: C = act(A@B + bias) [+ residual] ----------------
// A: MxK row-major, B: KxN row-major. One wave per 16x16 tile.
__global__ __launch_bounds__(32)
void gemm_kernel(const float* __restrict__ A, const float* __restrict__ B,
                 const float* __restrict__ bias, const float* __restrict__ residual,
                 float* __restrict__ C, int M, int N, int K, int act) {
  int lane = threadIdx.x;
  int half = lane >> 4, lh = lane & 15;
  int nb = blockIdx.x * 16, mb = blockIdx.y * 16;
  const float* arow = A + (size_t)(mb + lh) * K;
  v8f acc = {};
  for (int k0 = 0; k0 < K; k0 += 4) {
    int ka = k0 + 2 * half;
    v2f a, bf;
    a.x = arow[ka];
    a.y = arow[ka + 1];
    bf.x = B[(size_t)ka * N + nb + lh];
    bf.y = B[(size_t)(ka + 1) * N + nb + lh];
    acc = wmma_f32(a, bf, acc);
  }
  int n = nb + lh;
  float bc = bias ? bias[n] : 0.0f;
  for (int r = 0; r < 8; ++r) {
    int m = mb + r + 8 * half;
    float v = acc[r] + bc;
    if (act == 1) v = gelu_exact(v);
    if (residual) v += residual[(size_t)m * N + n];
    C[(size_t)m * N + n] = v;
  }
}

// ---------------- Fused attention: one wave per (16-query tile, head) ----------------
__global__ __launch_bounds__(32)
void attn_kernel(const float* __restrict__ q, const float* __restrict__ k,
                 const float* __restrict__ v, const unsigned char* __restrict__ mask,
                 float* __restrict__ ctx) {
  __shared__ float sc[16 * LSEQ];
  __shared__ float rmax[16], rsum[16];
  int lane = threadIdx.x;
  int half = lane >> 4, lh = lane & 15;
  int qbase = blockIdx.x * 16;
  int hoff = blockIdx.y * DHID;
  const float scale = 0.17677669529663687f;  // 1/sqrt(32)

  // ---- scores = q @ k^T, scaled + masked, staged to LDS ----
  for (int kt = 0; kt < LSEQ / 16; ++kt) {
    v8f acc = {};
    int key = kt * 16 + lh;
    for (int kk = 0; kk < DHID; kk += 4) {
      int ka = kk + 2 * half;
      v2f a, bf;
      a.x = q[(size_t)(qbase + lh) * DMODEL + hoff + ka];
      a.y = q[(size_t)(qbase + lh) * DMODEL + hoff + ka + 1];
      bf.x = k[(size_t)key * DMODEL + hoff + ka];
      bf.y = k[(size_t)key * DMODEL + hoff + ka + 1];
      acc = wmma_f32(a, bf, acc);
    }
    bool mk = mask[key] != 0;
    for (int r = 0; r < 8; ++r) {
      int m = r + 8 * half;
      sc[m * LSEQ + key] = mk ? acc[r] * scale : -1e9f;
    }
  }
  __syncthreads();

  // ---- softmax over 384 keys; lane pair (lh, half) splits each row ----
  {
    int row = lh, c0 = half * (LSEQ / 2);
    float mx = -INFINITY;
    for (int j = 0; j < LSEQ / 2; ++j) mx = fmaxf(mx, sc[row * LSEQ + c0 + j]);
    mx = fmaxf(mx, __shfl_xor(mx, 16, 32));
    float sum = 0.0f;
    for (int j = 0; j < LSEQ / 2; ++j) sum += expf(sc[row * LSEQ + c0 + j] - mx);
    sum += __shfl_xor(sum, 16, 32);
    if (half == 0) { rmax[row] = mx; rsum[row] = sum; }
  }
  __syncthreads();
  for (int idx = lane; idx < 16 * LSEQ; idx += 32) {
    int r2 = idx / LSEQ;
    sc[idx] = expf(sc[idx] - rmax[r2]) / rsum[r2];
  }
  __syncthreads();

  // ---- ctx = attn @ V ----
  for (int nt = 0; nt < 2; ++nt) {
    int col = hoff + nt * 16 + lh;
    v8f acc = {};
    for (int k0 = 0; k0 < LSEQ; k0 += 4) {
      int ka = k0 + 2 * half;
      v2f a, bf;
      a.x = sc[lh * LSEQ + ka];
      a.y = sc[lh * LSEQ + ka + 1];
      bf.x = v[(size_t)ka * DMODEL + col];
      bf.y = v[(size_t)(ka + 1) * DMODEL + col];
      acc = wmma_f32(a, bf, acc);
    }
    for (int r = 0; r < 8; ++r) {
      int m = qbase + r + 8 * half;
      ctx[(size_t)m * DMODEL + col] = acc[r];
    }
  }
}

// ---------------- Pair update (bandwidth kernel) ----------------
// pair_out[i,j,p] = pair_in[i,j,p] + (right @ T_i)[j,p] + bp[p]
// Block = 256 threads (8 waves) computes a 128(j) x 128(p) tile for one i.
// T_i (32x128 f32, 16KB) staged to LDS once per block via async-to-LDS b128.
// Each wave: 16 j-rows x 8 p-tiles -> 8 v8f accumulators, 64 WMMAs, A-frags
// (right) loaded once per k-step and reused across all 8 p-tiles.
__global__ __launch_bounds__(256)
void pair_update_kernel(const float* __restrict__ right, const float* __restrict__ T,
                        const float* __restrict__ pair_in, const float* __restrict__ bp,
                        float* __restrict__ pair_out) {
  __shared__ float ldsTi[DHID * DPAIR];  // [c*128 + p], 16KB
  int tid = threadIdx.x;
  int lane = tid & 31, wave = tid >> 5;
  int half = lane >> 4, lh = lane & 15;
  int i = blockIdx.y;
  int jb = blockIdx.x * 128 + wave * 16;
  const float* Ti = T + (size_t)i * (DHID * DPAIR);

  // stage T_i into LDS: 4096 floats, 32B (two b128) per lane per iteration
  for (int it = 0; it < 2; ++it) {
    int e = (it * 256 + tid) * 8;  // 8 floats = 2 x b128? no: use two b128 per thread
    async_copy_b128(&ldsTi[e], Ti + e);
    async_copy_b128(&ldsTi[e + 4], Ti + e + 4);
  }
  async_wait0();
  __syncthreads();

  v8f acc[8];
#pragma unroll
  for (int nt = 0; nt < 8; ++nt) acc[nt] = (v8f){};

  for (int k0 = 0; k0 < DHID; k0 += 4) {
    int ka = k0 + 2 * half;
    v2f a;
    a.x = right[(size_t)(jb + lh) * DHID + ka];
    a.y = right[(size_t)(jb + lh) * DHID + ka + 1];
#pragma unroll
    for (int nt = 0; nt < 8; ++nt) {
      v2f bf;
      bf.x = ldsTi[ka * DPAIR + nt * 16 + lh];
      bf.y = ldsTi[(ka + 1) * DPAIR + nt * 16 + lh];
      acc[nt] = wmma_f32(a, bf, acc[nt]);
    }
  }

  size_t base = (size_t)i * LSEQ * DPAIR;
#pragma unroll
  for (int nt = 0; nt < 8; ++nt) {
    int p = nt * 16 + lh;
    float bc = bp[p];
#pragma unroll
    for (int r = 0; r < 8; ++r) {
      int j = jb + r + 8 * half;
      size_t idx = base + (size_t)j * DPAIR + p;
      pair_out[idx] = pair_in[idx] + acc[nt][r] + bc;
    }
  }
}

// ---------------- Host orchestration ----------------
extern "C" void kernel_launch(void* const* d_in, const int* in_sizes, int n_in,
                              void* d_out, int out_size, void* d_ws, size_t ws_size,
                              hipStream_t stream) {
  (void)in_sizes; (void)n_in; (void)out_size; (void)ws_size;

  const float* s0        = (const float*)d_in[0];
  const float* pair_in   = (const float*)d_in[1];
  const unsigned char* mask = (const unsigned char*)d_in[2];
  const float* attn_ln_g = (const float*)d_in[3];
  const float* attn_ln_b = (const float*)d_in[4];
  const float* Wq = (const float*)d_in[5];  const float* bq = (const float*)d_in[6];
  const float* Wk = (const float*)d_in[7];  const float* bk = (const float*)d_in[8];
  const float* Wv = (const float*)d_in[9];  const float* bv = (const float*)d_in[10];
  const float* Wo = (const float*)d_in[11]; const float* bo = (const float*)d_in[12];
  const float* pu_ln_g = (const float*)d_in[13];
  const float* pu_ln_b = (const float*)d_in[14];
  const float* Wl = (const float*)d_in[15]; const float* bl = (const float*)d_in[16];
  const float* Wr = (const float*)d_in[17]; const float* br = (const float*)d_in[18];
  const float* Wp = (const float*)d_in[19]; const float* bp = (const float*)d_in[20];
  const float* tr_ln_g = (const float*)d_in[21];
  const float* tr_ln_b = (const float*)d_in[22];
  const float* W1 = (const float*)d_in[23]; const float* b1 = (const float*)d_in[24];
  const float* W2 = (const float*)d_in[25]; const float* b2 = (const float*)d_in[26];

  float* ws = (float*)d_ws;
  const size_t SM = (size_t)LSEQ * DMODEL;      // 98304
  float* h     = ws;
  float* q     = ws + SM;
  float* k     = ws + 2 * SM;
  float* v     = ws + 3 * SM;
  float* ctx   = ws + 4 * SM;
  float* s1    = ws + 5 * SM;
  float* left  = ws + 6 * SM;                   // 384*32
  float* right = left + (size_t)LSEQ * DHID;
  float* Tbuf  = right + (size_t)LSEQ * DHID;   // 384*4096
  float* ffn1  = Tbuf + (size_t)LSEQ * DHID * DPAIR;  // 384*1024

  float* out_s    = (float*)d_out;
  float* out_pair = out_s + SM;

  dim3 w32(32);

  // --- attention ---
  ln_kernel<<<dim3(LSEQ), dim3(DMODEL), 0, stream>>>(s0, attn_ln_g, attn_ln_b, h);
  gemm_kernel<<<dim3(DMODEL/16, LSEQ/16), w32, 0, stream>>>(h, Wq, bq, nullptr, q,
                                                            LSEQ, DMODEL, DMODEL, 0);
  gemm_kernel<<<dim3(DMODEL/16, LSEQ/16), w32, 0, stream>>>(h, Wk, bk, nullptr, k,
                                                            LSEQ, DMODEL, DMODEL, 0);
  gemm_kernel<<<dim3(DMODEL/16, LSEQ/16), w32, 0, stream>>>(h, Wv, bv, nullptr, v,
                                                            LSEQ, DMODEL, DMODEL, 0);
  attn_kernel<<<dim3(LSEQ/16, 8), w32, 0, stream>>>(q, k, v, mask, ctx);
  gemm_kernel<<<dim3(DMODEL/16, LSEQ/16), w32, 0, stream>>>(ctx, Wo, bo, s0, s1,
                                                            LSEQ, DMODEL, DMODEL, 0);

  // --- pair update (factorized bilinear) ---
  ln_kernel<<<dim3(LSEQ), dim3(DMODEL), 0, stream>>>(s1, pu_ln_g, pu_ln_b, h);
  gemm_kernel<<<dim3(DHID/16, LSEQ/16), w32, 0, stream>>>(h, Wl, bl, nullptr, left,
                                                          LSEQ, DHID, DMODEL, 0);
  gemm_kernel<<<dim3(DHID/16, LSEQ/16), w32, 0, stream>>>(h, Wr, br, nullptr, right,
                                                          LSEQ, DHID, DMODEL, 0);
  gemm_kernel<<<dim3((DHID*DPAIR)/16, LSEQ/16), w32, 0, stream>>>(
      left, Wp, nullptr, nullptr, Tbuf, LSEQ, DHID * DPAIR, DHID, 0);
  pair_update_kernel<<<dim3(LSEQ/128, LSEQ), dim3(256), 0, stream>>>(
      right, Tbuf, pair_in, bp, out_pair);

  // --- transition FFN ---
  ln_kernel<<<dim3(LSEQ), dim3(DMODEL), 0, stream>>>(s1, tr_ln_g, tr_ln_b, h);
  gemm_kernel<<<dim3(DFF/16, LSEQ/16), w32, 0, stream>>>(h, W1, b1, nullptr, ffn1,
                                                         LSEQ, DFF, DMODEL, 1);
  gemm_kernel<<<dim3(DMODEL/16, LSEQ/16), w32, 0, stream>>>(ffn1, W2, b2, s1, out_s,
                                                            LSEQ, DMODEL, DFF, 0);
}